// block_10737418240064
// MI455X (gfx1250) — compile-verified
//
#include <hip/hip_runtime.h>

// ---------------- problem constants ----------------
constexpr int Bsz  = 2;
constexpr int Cc   = 96;
constexpr int C2c  = 192;
constexpr int Hh   = 256;
constexpr int Wh   = 256;
constexpr int HWc  = Hh * Wh;            // 65536
constexpr int HPp  = 260;
constexpr int WPp  = 260;
constexpr int NPIXc = HPp * WPp;         // 67600
constexpr int NIMGc = Bsz * C2c;         // 384 images through the FFT filter
constexpr int MROWSc = NIMGc * HPp;      // 99840 rows for the row-transforms

// ---------------- workspace layout (float offsets) ----------------
constexpr size_t OFF_CTAB = 0;                               // cos(2*pi*a*b/260), 260x260
constexpr size_t OFF_STAB = OFF_CTAB + NPIXc;                // sin table
constexpr size_t OFF_WK   = OFF_STAB + NPIXc;                // softmaxed 3x3 kernels, 192*9
constexpr size_t OFF_SIG  = OFF_WK + (size_t)C2c * 9;        // per-channel sigma, 192
constexpr size_t OFF_XN   = (OFF_SIG + C2c + 15) & ~(size_t)15;  // LN1 out; later x2
constexpr size_t N_XN     = (size_t)Bsz * Cc * HWc;
constexpr size_t OFF_H1   = OFF_XN + N_XN;                   // h1; later h2, h3
constexpr size_t N_H1     = (size_t)Bsz * C2c * HWc;
constexpr size_t OFF_G    = OFF_H1 + N_H1;                   // padded images; later xn2
constexpr size_t N_G      = (size_t)NIMGc * NPIXc;
constexpr size_t OFF_T    = OFF_G + N_G;                     // T planes (Tr,Ti); later U
constexpr size_t N_PLANE  = (size_t)MROWSc * WPp;            // 25,958,400
constexpr size_t OFF_GF   = OFF_T + 2 * N_PLANE;             // G planes (Gr,Gi)

// ---------------- vector types / helpers ----------------
typedef __attribute__((ext_vector_type(2))) float v2f;
typedef __attribute__((ext_vector_type(8))) float v8f;

__device__ __forceinline__ v8f v8zero() {
  v8f z = {0.f, 0.f, 0.f, 0.f, 0.f, 0.f, 0.f, 0.f};
  return z;
}

// D(16x16,f32) = A(16x4,f32) * B(4x16,f32) + C
__device__ __forceinline__ v8f wmma4(v2f a, v2f b, v8f c) {
  return __builtin_amdgcn_wmma_f32_16x16x4_f32(false, a, false, b, (short)0, c,
                                               false, false);
}

__device__ __forceinline__ float gelu_f(float x) {
  return 0.5f * x * (1.0f + erff(x * 0.7071067811865476f));
}

__device__ __forceinline__ int imin(int a, int b) { return a < b ? a : b; }
__device__ __forceinline__ int imax(int a, int b) { return a > b ? a : b; }

// ---------------- one-time-per-launch precompute ----------------
__global__ void k_tables(float* __restrict__ Ct, float* __restrict__ St) {
  int t = blockIdx.x * blockDim.x + threadIdx.x;
  if (t >= NPIXc) return;
  int a = t / WPp, b = t % WPp;
  int m = (a * b) % 260;  // exact integer reduction
  double ang = (6.283185307179586476925286766559 / 260.0) * (double)m;
  Ct[t] = (float)cos(ang);
  St[t] = (float)sin(ang);
}

__global__ void k_chanprep(const float* __restrict__ cw, const float* __restrict__ cs,
                           float* __restrict__ wk, float* __restrict__ sig) {
  int c = blockIdx.x * blockDim.x + threadIdx.x;
  if (c >= C2c) return;
  float v[9];
  float mx = -1e30f;
  for (int j = 0; j < 9; j++) { v[j] = cw[c * 9 + j]; mx = fmaxf(mx, v[j]); }
  float s = 0.f;
  for (int j = 0; j < 9; j++) { v[j] = expf(v[j] - mx); s += v[j]; }
  float inv = 1.0f / s;
  for (int j = 0; j < 9; j++) wk[c * 9 + j] = v[j] * inv;
  float sp = cs[c];
  sig[c] = 1.0f / (1.0f + expf(-(sp - 9.0f))) + 1e-5f;
}

// ---------------- channels-first LayerNorm (per pixel over 96 channels) -------
__global__ void k_ln(const float* __restrict__ x, const float* __restrict__ g,
                     const float* __restrict__ b, float* __restrict__ out) {
  int p = blockIdx.x * blockDim.x + threadIdx.x;
  if (p >= Bsz * HWc) return;
  int bb = p / HWc, hw = p % HWc;
  const float* xp = x + (size_t)bb * Cc * HWc + hw;
  float s = 0.f, s2 = 0.f;
  for (int c = 0; c < Cc; c++) {
    float v = xp[(size_t)c * HWc];
    s += v; s2 += v * v;
  }
  float mu = s * (1.0f / Cc);
  float var = s2 * (1.0f / Cc) - mu * mu;
  float rinv = rsqrtf(var + 1e-5f);
  float* op = out + (size_t)bb * Cc * HWc + hw;
  for (int c = 0; c < Cc; c++) {
    float v = xp[(size_t)c * HWc];
    op[(size_t)c * HWc] = g[c] * ((v - mu) * rinv) + b[c];
  }
}

// ---------------- replicate pad 256x256 -> 260x260 ----------------
__global__ void k_pad(const float* __restrict__ h, float* __restrict__ g) {
  int t = blockIdx.x * blockDim.x + threadIdx.x;
  if (t >= NIMGc * NPIXc) return;
  int img = t / NPIXc, rem = t % NPIXc;
  int y = rem / WPp, x = rem % WPp;
  int ys = imin(imax(y - 2, 0), Hh - 1);
  int xs = imin(imax(x - 2, 0), Wh - 1);
  g[t] = h[(size_t)img * HWc + (size_t)ys * Wh + xs];
}

// ---------------- 1x1 conv as WMMA GEMM: out[m,n] = act(W[m,:].in[:,n]+b)(+res)
template <int M, int K, bool ACT, bool RES>
__global__ __launch_bounds__(256) void k_gemm_pw(
    const float* __restrict__ in, const float* __restrict__ Wt,
    const float* __restrict__ bias, const float* __restrict__ res,
    float* __restrict__ out) {
  constexpr int MT = M / 16;
  constexpr int NT = HWc / 64;
  int wave = (blockIdx.x * blockDim.x + threadIdx.x) >> 5;
  if (wave >= Bsz * MT * NT) return;
  int nT = wave % NT;
  int mT = (wave / NT) % MT;
  int bb = wave / (NT * MT);
  int lane = threadIdx.x & 31, lm = lane & 15, lk = lane >> 4;

  const float* inb = in + (size_t)bb * K * HWc;
  const float* wp  = Wt + (size_t)(mT * 16 + lm) * K + 2 * lk;  // A: row lm, cols kA..
  const float* rb0[4];
  const float* rb1[4];
#pragma unroll
  for (int s = 0; s < 4; s++) {
    rb0[s] = inb + (size_t)(2 * lk) * HWc + nT * 64 + s * 16 + lm;  // row kA
    rb1[s] = rb0[s] + HWc;                                          // row kA+1
  }

  v8f acc[4] = {v8zero(), v8zero(), v8zero(), v8zero()};
  for (int k0 = 0; k0 < K; k0 += 4) {
    v2f a = *(const v2f*)(wp + k0);
    size_t ko = (size_t)k0 * HWc;
#pragma unroll
    for (int s = 0; s < 4; s++) {
      v2f bf;
      bf.x = rb0[s][ko];
      bf.y = rb1[s][ko];
      acc[s] = wmma4(a, bf, acc[s]);
    }
  }
  float* ob = out + (size_t)bb * M * HWc;
#pragma unroll
  for (int s = 0; s < 4; s++) {
#pragma unroll
    for (int i = 0; i < 8; i++) {
      int m = mT * 16 + i + 8 * lk;
      int n = nT * 64 + s * 16 + lm;
      float v = acc[s][i] + bias[m];
      if (ACT) v = gelu_f(v);
      if (RES) v += res[(size_t)bb * M * HWc + (size_t)m * HWc + n];
      ob[(size_t)m * HWc + n] = v;
    }
  }
}

// ---------------- forward DFT over rows (x -> u): T[y,u] = sum_x g[y,x] e^{-i2pi ux/260}
// B-fragments come from the symmetric tables: Ct[k*260+n] == Ct[n*260+k] -> v2f loads.
__global__ __launch_bounds__(256) void k_dft_rows_fwd(
    const float* __restrict__ g, const float* __restrict__ Ct,
    const float* __restrict__ St, float* __restrict__ Tr, float* __restrict__ Ti) {
  constexpr int MT = MROWSc / 16;  // 6240
  constexpr int NT = 9;            // 9*32 >= 260
  int wave = (blockIdx.x * blockDim.x + threadIdx.x) >> 5;
  if (wave >= MT * NT) return;
  int nT = wave % NT, mT = wave / NT;
  int lane = threadIdx.x & 31, lm = lane & 15, lk = lane >> 4;

  const float* ap = g + (size_t)(mT * 16 + lm) * WPp + 2 * lk;
  const float* pc[2];
  const float* ps[2];
#pragma unroll
  for (int s = 0; s < 2; s++) {
    int nc = imin(nT * 32 + s * 16 + lm, WPp - 1);  // clamped: spills only into
    pc[s] = Ct + (size_t)nc * WPp + 2 * lk;         // never-stored output columns
    ps[s] = St + (size_t)nc * WPp + 2 * lk;
  }

  v8f aC[2] = {v8zero(), v8zero()};
  v8f aS[2] = {v8zero(), v8zero()};
#pragma unroll 5
  for (int k0 = 0; k0 < WPp; k0 += 4) {
    v2f a = *(const v2f*)(ap + k0);
#pragma unroll
    for (int s = 0; s < 2; s++) {
      v2f bc = *(const v2f*)(pc[s] + k0);
      v2f bs = *(const v2f*)(ps[s] + k0);
      aC[s] = wmma4(a, bc, aC[s]);
      aS[s] = wmma4(a, bs, aS[s]);
    }
  }
#pragma unroll
  for (int s = 0; s < 2; s++) {
    int n = nT * 32 + s * 16 + lm;
    if (n < WPp) {
#pragma unroll
      for (int i = 0; i < 8; i++) {
        int m = mT * 16 + i + 8 * lk;
        Tr[(size_t)m * WPp + n] = aC[s][i];
        Ti[(size_t)m * WPp + n] = -aS[s][i];
      }
    }
  }
}

// ---------------- forward DFT over cols (y -> v): G = (C - iS) x T ----------------
// 32 output rows per wave: B fragments (streamed T planes) shared by both row sets.
__global__ __launch_bounds__(256) void k_dft_cols_fwd(
    const float* __restrict__ Tr, const float* __restrict__ Ti,
    const float* __restrict__ Ct, const float* __restrict__ St,
    float* __restrict__ Gr, float* __restrict__ Gi) {
  constexpr int MT = 9, NT = 9;  // 9*32 >= 260 rows, 9*32 >= 260 cols
  int wave = (blockIdx.x * blockDim.x + threadIdx.x) >> 5;
  if (wave >= NIMGc * MT * NT) return;
  int nT = wave % NT;
  int mT = (wave / NT) % MT;
  int img = wave / (NT * MT);
  int lane = threadIdx.x & 31, lm = lane & 15, lk = lane >> 4;

  int v0 = imin(mT * 32 + lm, WPp - 1);       // clamped A rows -> unstored rows
  int v1 = imin(mT * 32 + 16 + lm, WPp - 1);
  const float* crow0 = Ct + (size_t)v0 * WPp + 2 * lk;
  const float* srow0 = St + (size_t)v0 * WPp + 2 * lk;
  const float* crow1 = Ct + (size_t)v1 * WPp + 2 * lk;
  const float* srow1 = St + (size_t)v1 * WPp + 2 * lk;

  const float* tr0[2];
  const float* tr1[2];
  const float* ti0[2];
  const float* ti1[2];
#pragma unroll
  for (int s = 0; s < 2; s++) {
    int nc = imin(nT * 32 + s * 16 + lm, WPp - 1);
    const float* base = Tr + (size_t)img * NPIXc + (size_t)(2 * lk) * WPp + nc;
    tr0[s] = base;
    tr1[s] = base + WPp;
    const float* basei = Ti + (size_t)img * NPIXc + (size_t)(2 * lk) * WPp + nc;
    ti0[s] = basei;
    ti1[s] = basei + WPp;
  }

  v8f gr[2][2] = {{v8zero(), v8zero()}, {v8zero(), v8zero()}};
  v8f gi[2][2] = {{v8zero(), v8zero()}, {v8zero(), v8zero()}};
#pragma unroll 5
  for (int k0 = 0; k0 < HPp; k0 += 4) {
    v2f aC0 = *(const v2f*)(crow0 + k0);
    v2f aS0 = *(const v2f*)(srow0 + k0);
    v2f aC1 = *(const v2f*)(crow1 + k0);
    v2f aS1 = *(const v2f*)(srow1 + k0);
    v2f aSn0, aSn1;
    aSn0.x = -aS0.x; aSn0.y = -aS0.y;
    aSn1.x = -aS1.x; aSn1.y = -aS1.y;
    size_t ko = (size_t)k0 * WPp;
#pragma unroll
    for (int s = 0; s < 2; s++) {
      v2f btr, bti;
      btr.x = tr0[s][ko];
      btr.y = tr1[s][ko];
      bti.x = ti0[s][ko];
      bti.y = ti1[s][ko];
      gr[0][s] = wmma4(aC0, btr, gr[0][s]);   // + C*Tr
      gr[0][s] = wmma4(aS0, bti, gr[0][s]);   // + S*Ti
      gi[0][s] = wmma4(aC0, bti, gi[0][s]);   // + C*Ti
      gi[0][s] = wmma4(aSn0, btr, gi[0][s]);  // - S*Tr
      gr[1][s] = wmma4(aC1, btr, gr[1][s]);
      gr[1][s] = wmma4(aS1, bti, gr[1][s]);
      gi[1][s] = wmma4(aC1, bti, gi[1][s]);
      gi[1][s] = wmma4(aSn1, btr, gi[1][s]);
    }
  }
#pragma unroll
  for (int r = 0; r < 2; r++) {
#pragma unroll
    for (int s = 0; s < 2; s++) {
      int n = nT * 32 + s * 16 + lm;
#pragma unroll
      for (int i = 0; i < 8; i++) {
        int m = mT * 32 + r * 16 + i + 8 * lk;  // = v
        if (m < HPp && n < WPp) {
          size_t o = ((size_t)img * HPp + m) * WPp + n;
          Gr[o] = gr[r][s][i];
          Gi[o] = gi[r][s][i];
        }
      }
    }
  }
}

// ---------------- per-channel transfer function, in place on G ----------------
__global__ void k_filter(float* __restrict__ Gr, float* __restrict__ Gi,
                         const float* __restrict__ Ct, const float* __restrict__ St,
                         const float* __restrict__ wk, const float* __restrict__ sig) {
  int t = blockIdx.x * blockDim.x + threadIdx.x;
  if (t >= NIMGc * NPIXc) return;
  int img = t / NPIXc, rem = t % NPIXc;
  int v = rem / WPp, u = rem % WPp;
  int c = img % C2c;
  float fbr = 0.f, fbi = 0.f;
#pragma unroll
  for (int i = 0; i < 3; i++) {
#pragma unroll
    for (int j = 0; j < 3; j++) {
      int mm = (i - 1) * v + (j - 1) * u;
      mm %= 260;
      if (mm < 0) mm += 260;
      float wv = wk[c * 9 + i * 3 + j];
      // e^{-i*2pi*mm/260} = cos - i sin ; row 1 of the tables is the angle table
      fbr += wv * Ct[260 + mm];
      fbi -= wv * St[260 + mm];
    }
  }
  float sg = sig[c];
  float d = 1.0f / (fbr * fbr + fbi * fbi + sg);
  float hr = (fbr + sg) * d;
  float hi = -fbi * d;
  float grv = Gr[t], giv = Gi[t];
  Gr[t] = hr * grv - hi * giv;
  Gi[t] = hr * giv + hi * grv;
}

// ---------------- inverse DFT over rows (u -> x): U = G' x (C + iS) ----------------
__global__ __launch_bounds__(256) void k_dft_rows_inv(
    const float* __restrict__ Gr, const float* __restrict__ Gi,
    const float* __restrict__ Ct, const float* __restrict__ St,
    float* __restrict__ Ur, float* __restrict__ Ui) {
  constexpr int MT = MROWSc / 16;
  constexpr int NT = 9;
  int wave = (blockIdx.x * blockDim.x + threadIdx.x) >> 5;
  if (wave >= MT * NT) return;
  int nT = wave % NT, mT = wave / NT;
  int lane = threadIdx.x & 31, lm = lane & 15, lk = lane >> 4;

  const float* ar = Gr + (size_t)(mT * 16 + lm) * WPp + 2 * lk;
  const float* ai = Gi + (size_t)(mT * 16 + lm) * WPp + 2 * lk;
  const float* pc[2];
  const float* ps[2];
#pragma unroll
  for (int s = 0; s < 2; s++) {
    int nc = imin(nT * 32 + s * 16 + lm, WPp - 1);
    pc[s] = Ct + (size_t)nc * WPp + 2 * lk;  // symmetric-table v2f access
    ps[s] = St + (size_t)nc * WPp + 2 * lk;
  }

  v8f ur[2] = {v8zero(), v8zero()};
  v8f ui[2] = {v8zero(), v8zero()};
#pragma unroll 5
  for (int k0 = 0; k0 < WPp; k0 += 4) {
    v2f aR = *(const v2f*)(ar + k0);
    v2f aI = *(const v2f*)(ai + k0);
    v2f aIn;
    aIn.x = -aI.x;
    aIn.y = -aI.y;
#pragma unroll
    for (int s = 0; s < 2; s++) {
      v2f bc = *(const v2f*)(pc[s] + k0);
      v2f bs = *(const v2f*)(ps[s] + k0);
      ur[s] = wmma4(aR, bc, ur[s]);   // + Gr*C
      ur[s] = wmma4(aIn, bs, ur[s]);  // - Gi*S
      ui[s] = wmma4(aR, bs, ui[s]);   // + Gr*S
      ui[s] = wmma4(aI, bc, ui[s]);   // + Gi*C
    }
  }
#pragma unroll
  for (int s = 0; s < 2; s++) {
    int n = nT * 32 + s * 16 + lm;
    if (n < WPp) {
#pragma unroll
      for (int i = 0; i < 8; i++) {
        int m = mT * 16 + i + 8 * lk;
        Ur[(size_t)m * WPp + n] = ur[s][i];
        Ui[(size_t)m * WPp + n] = ui[s][i];
      }
    }
  }
}

// ------ inverse DFT over cols + 1/N + crop + GELU: h2[y-2,x-2] = gelu(real/N) ------
// 32 output rows per wave: B fragments (streamed U planes) shared by both row sets.
__global__ __launch_bounds__(256) void k_dft_cols_inv(
    const float* __restrict__ Ur, const float* __restrict__ Ui,
    const float* __restrict__ Ct, const float* __restrict__ St,
    float* __restrict__ h2) {
  constexpr int MT = 9, NT = 9;
  int wave = (blockIdx.x * blockDim.x + threadIdx.x) >> 5;
  if (wave >= NIMGc * MT * NT) return;
  int nT = wave % NT;
  int mT = (wave / NT) % MT;
  int img = wave / (NT * MT);
  int lane = threadIdx.x & 31, lm = lane & 15, lk = lane >> 4;

  int y0 = imin(mT * 32 + lm, HPp - 1);
  int y1 = imin(mT * 32 + 16 + lm, HPp - 1);
  const float* crow0 = Ct + (size_t)y0 * WPp + 2 * lk;
  const float* srow0 = St + (size_t)y0 * WPp + 2 * lk;
  const float* crow1 = Ct + (size_t)y1 * WPp + 2 * lk;
  const float* srow1 = St + (size_t)y1 * WPp + 2 * lk;

  const float* ur0[2];
  const float* ur1[2];
  const float* ui0[2];
  const float* ui1[2];
#pragma unroll
  for (int s = 0; s < 2; s++) {
    int nc = imin(nT * 32 + s * 16 + lm, WPp - 1);
    const float* base = Ur + (size_t)img * NPIXc + (size_t)(2 * lk) * WPp + nc;
    ur0[s] = base;
    ur1[s] = base + WPp;
    const float* basei = Ui + (size_t)img * NPIXc + (size_t)(2 * lk) * WPp + nc;
    ui0[s] = basei;
    ui1[s] = basei + WPp;
  }

  v8f acc[2][2] = {{v8zero(), v8zero()}, {v8zero(), v8zero()}};
#pragma unroll 5
  for (int k0 = 0; k0 < HPp; k0 += 4) {
    v2f aC0 = *(const v2f*)(crow0 + k0);
    v2f aS0 = *(const v2f*)(srow0 + k0);
    v2f aC1 = *(const v2f*)(crow1 + k0);
    v2f aS1 = *(const v2f*)(srow1 + k0);
    v2f aSn0, aSn1;
    aSn0.x = -aS0.x; aSn0.y = -aS0.y;
    aSn1.x = -aS1.x; aSn1.y = -aS1.y;
    size_t ko = (size_t)k0 * WPp;
#pragma unroll
    for (int s = 0; s < 2; s++) {
      v2f bur, bui;
      bur.x = ur0[s][ko];
      bur.y = ur1[s][ko];
      bui.x = ui0[s][ko];
      bui.y = ui1[s][ko];
      acc[0][s] = wmma4(aC0, bur, acc[0][s]);   // + C*Ur
      acc[0][s] = wmma4(aSn0, bui, acc[0][s]);  // - S*Ui
      acc[1][s] = wmma4(aC1, bur, acc[1][s]);
      acc[1][s] = wmma4(aSn1, bui, acc[1][s]);
    }
  }
  constexpr float invN = 1.0f / (float)NPIXc;
#pragma unroll
  for (int r = 0; r < 2; r++) {
#pragma unroll
    for (int s = 0; s < 2; s++) {
      int n = nT * 32 + s * 16 + lm;
#pragma unroll
      for (int i = 0; i < 8; i++) {
        int m = mT * 32 + r * 16 + i + 8 * lk;  // = y
        if (m >= 2 && m < HPp - 2 && n >= 2 && n < WPp - 2) {
          h2[(size_t)img * HWc + (size_t)(m - 2) * Wh + (n - 2)] =
              gelu_f(acc[r][s][i] * invN);
        }
      }
    }
  }
}

// ---------------- launcher ----------------
extern "C" void kernel_launch(void* const* d_in, const int* in_sizes, int n_in,
                              void* d_out, int out_size, void* d_ws, size_t ws_size,
                              hipStream_t stream) {
  const float* x         = (const float*)d_in[0];
  const float* ln1_w     = (const float*)d_in[1];
  const float* ln1_b     = (const float*)d_in[2];
  const float* w1a       = (const float*)d_in[3];
  const float* b1a       = (const float*)d_in[4];
  const float* conv_w    = (const float*)d_in[5];
  const float* conv_sig  = (const float*)d_in[6];
  const float* w1b       = (const float*)d_in[7];
  const float* b1b       = (const float*)d_in[8];
  const float* ln2_w     = (const float*)d_in[9];
  const float* ln2_b     = (const float*)d_in[10];
  const float* w2a       = (const float*)d_in[11];
  const float* b2a       = (const float*)d_in[12];
  const float* w2b       = (const float*)d_in[13];
  const float* b2b       = (const float*)d_in[14];

  float* ws   = (float*)d_ws;
  float* Ctab = ws + OFF_CTAB;
  float* Stab = ws + OFF_STAB;
  float* wk   = ws + OFF_WK;
  float* sig  = ws + OFF_SIG;
  float* xn   = ws + OFF_XN;   // also x2
  float* h1   = ws + OFF_H1;   // also h2, h3
  float* g    = ws + OFF_G;    // also xn2
  float* Tr   = ws + OFF_T;    // also Ur
  float* Ti   = Tr + N_PLANE;  // also Ui
  float* Gr   = ws + OFF_GF;
  float* Gi   = Gr + N_PLANE;

  dim3 blk(256);

  k_tables<<<(NPIXc + 255) / 256, blk, 0, stream>>>(Ctab, Stab);
  k_chanprep<<<1, blk, 0, stream>>>(conv_w, conv_sig, wk, sig);

  // branch 1: LN -> expand+GELU -> converse2d -> GELU -> project + residual
  k_ln<<<(Bsz * HWc) / 256, blk, 0, stream>>>(x, ln1_w, ln1_b, xn);
  k_gemm_pw<C2c, Cc, true, false>
      <<<(Bsz * (C2c / 16) * (HWc / 64)) / 8, blk, 0, stream>>>(xn, w1a, b1a, nullptr, h1);
  k_pad<<<(NIMGc * NPIXc) / 256, blk, 0, stream>>>(h1, g);
  k_dft_rows_fwd<<<((MROWSc / 16) * 9) / 8, blk, 0, stream>>>(g, Ctab, Stab, Tr, Ti);
  k_dft_cols_fwd<<<(NIMGc * 9 * 9) / 8, blk, 0, stream>>>(Tr, Ti, Ctab, Stab, Gr, Gi);
  k_filter<<<(NIMGc * NPIXc) / 256, blk, 0, stream>>>(Gr, Gi, Ctab, Stab, wk, sig);
  k_dft_rows_inv<<<((MROWSc / 16) * 9) / 8, blk, 0, stream>>>(Gr, Gi, Ctab, Stab, Tr, Ti);
  k_dft_cols_inv<<<(NIMGc * 9 * 9) / 8, blk, 0, stream>>>(Tr, Ti, Ctab, Stab, h1);
  k_gemm_pw<Cc, C2c, false, true>
      <<<(Bsz * (Cc / 16) * (HWc / 64)) / 8, blk, 0, stream>>>(h1, w1b, b1b, x, xn);

  // branch 2: LN -> expand+GELU -> project + residual
  k_ln<<<(Bsz * HWc) / 256, blk, 0, stream>>>(xn, ln2_w, ln2_b, g);
  k_gemm_pw<C2c, Cc, true, false>
      <<<(Bsz * (C2c / 16) * (HWc / 64)) / 8, blk, 0, stream>>>(g, w2a, b2a, nullptr, h1);
  k_gemm_pw<Cc, C2c, false, true>
      <<<(Bsz * (Cc / 16) * (HWc / 64)) / 8, blk, 0, stream>>>(h1, w2b, b2b, xn,
                                                               (float*)d_out);
}